// Custom_RoPE_13228499272230
// MI455X (gfx1250) — compile-verified
//
#include <hip/hip_runtime.h>
#include <stdint.h>

typedef float v4f __attribute__((ext_vector_type(4)));

// ds_swizzle_b32 bitmask mode: offset = {1'b0, xor[14:10], or[9:5], and[4:0]}
template <int XORMASK>
__device__ __forceinline__ float swz_xor(float v) {
    return __int_as_float(
        __builtin_amdgcn_ds_swizzle(__float_as_int(v), (XORMASK << 10) | 0x1f));
}

// Grid: x = T/8 tiles of 8 timesteps, y = H heads. Block = 256 = 8 waves,
// one wave32 per (h,t) row of 128 floats.
__global__ __launch_bounds__(256)
void rope_obs_kernel(const float* __restrict__ x,
                     const float* __restrict__ cosp,
                     const float* __restrict__ sinp,
                     float* __restrict__ out,
                     float* __restrict__ omax,
                     float* __restrict__ omin,
                     int T)
{
    // 8 rows * 128 floats for cos, then same for sin: 8 KB LDS
    __shared__ __align__(16) float ldsbuf[2048];

    const int tid  = threadIdx.x;
    const int lane = tid & 31;
    const int wave = tid >> 5;

    const int t0 = blockIdx.x * 8;                        // first timestep of tile

    // ---- Async copy of this block's cos/sin tiles into LDS (CDNA5 async path).
    // Each of the 256 threads moves one 16B chunk per table (256*16B = 4KB each).
    {
        const float* cg = cosp + (size_t)t0 * 128;
        const float* sg = sinp + (size_t)t0 * 128;
        // Low 32 bits of a generic pointer into LDS == workgroup-relative LDS
        // byte offset (LDS aperture keeps the offset in addr[31:0]).
        unsigned ldsC = (unsigned)(uintptr_t)(&ldsbuf[0])    + (unsigned)tid * 16u;
        unsigned ldsS = (unsigned)(uintptr_t)(&ldsbuf[1024]) + (unsigned)tid * 16u;
        unsigned voff = (unsigned)tid * 16u;
        asm volatile("global_load_async_to_lds_b128 %0, %1, %2\n\t"
                     "global_load_async_to_lds_b128 %3, %1, %4"
                     :
                     : "v"(ldsC), "v"(voff), "s"(cg), "v"(ldsS), "s"(sg)
                     : "memory");
        asm volatile("s_wait_asynccnt 0x0" ::: "memory");
    }
    __syncthreads();

    // ---- One wave per row; lane l owns elements 4l..4l+3 (one b128).
    const size_t r    = (size_t)blockIdx.y * (size_t)T + (size_t)(t0 + wave);
    const float* xrow = x   + r * 128;
    float*       orow = out + r * 128;

    // x / out are pure streams: non-temporal so the 192MB L2 keeps cos/sin hot.
    v4f xv = __builtin_nontemporal_load(((const v4f*)xrow) + lane);

    const int base = wave * 128 + lane * 4;
    v4f cv = *(const v4f*)(&ldsbuf[base]);
    v4f sv = *(const v4f*)(&ldsbuf[1024 + base]);

    // rotate_half partner: element e^64 lives in lane l^16, same component.
    v4f xp;
    xp.x = swz_xor<16>(xv.x);
    xp.y = swz_xor<16>(xv.y);
    xp.z = swz_xor<16>(xv.z);
    xp.w = swz_xor<16>(xv.w);
    const float sgn = (lane < 16) ? -1.0f : 1.0f;   // lanes 0-15 hold -x2 terms

    v4f o = xv * cv + (sgn * xp) * sv;

    __builtin_nontemporal_store(o, ((v4f*)orow) + lane);

    // ---- per-row max/min: lane-local 4-way, then wave32 xor-swizzle tree.
    float mx = fmaxf(fmaxf(o.x, o.y), fmaxf(o.z, o.w));
    float mn = fminf(fminf(o.x, o.y), fminf(o.z, o.w));

    mx = fmaxf(mx, swz_xor<16>(mx));  mn = fminf(mn, swz_xor<16>(mn));
    mx = fmaxf(mx, swz_xor<8>(mx));   mn = fminf(mn, swz_xor<8>(mn));
    mx = fmaxf(mx, swz_xor<4>(mx));   mn = fminf(mn, swz_xor<4>(mn));
    mx = fmaxf(mx, swz_xor<2>(mx));   mn = fminf(mn, swz_xor<2>(mn));
    mx = fmaxf(mx, swz_xor<1>(mx));   mn = fminf(mn, swz_xor<1>(mn));

    if (lane == 0) {
        omax[r] = mx;
        omin[r] = mn;
    }
}

extern "C" void kernel_launch(void* const* d_in, const int* in_sizes, int n_in,
                              void* d_out, int out_size, void* d_ws, size_t ws_size,
                              hipStream_t stream) {
    // setup_inputs order: x, scale_x, cos, scale_cos, sin, scale_sin
    // (scale_x / scale_cos / scale_sin are unused by the reference)
    const float* x    = (const float*)d_in[0];
    const float* cosp = (const float*)d_in[2];
    const float* sinp = (const float*)d_in[4];

    const long long HT = in_sizes[1];              // H*T = 262144
    const int       D  = (int)(in_sizes[0] / HT);  // 128
    const int       T  = (int)(in_sizes[2] / D);   // 8192
    const int       H  = (int)(HT / T);            // 32

    float* out  = (float*)d_out;
    float* omax = out + HT * (long long)D;
    float* omin = omax + HT;

    dim3 grid(T / 8, H, 1);                        // 1024 x 32 blocks
    rope_obs_kernel<<<grid, 256, 0, stream>>>(x, cosp, sinp, out, omax, omin, T);
}